// quantizer_53446573032170
// MI455X (gfx1250) — compile-verified
//
#include <hip/hip_runtime.h>

typedef __attribute__((ext_vector_type(2))) float v2f;
typedef __attribute__((ext_vector_type(8))) float v8f;

#define BATCH 32
#define LATD  32
#define HW    1024               // H*W
#define KCB   8192
#define NTOK  (BATCH * HW)       // 32768 tokens
#define CHUNK 256                // codebook codes staged in LDS per iteration
#define TPB   256                // 8 wave32s
#define LDS_STRIDE 34            // 32 floats + 2 pad (keeps 8B align, spreads banks)
#define Q_ELEMS (BATCH * LATD * HW)   // 1048576 quant_out floats
#define I_ELEMS NTOK                  // 32768 index outputs (stored as float)

// Each block: 128 tokens (8 waves x 16-token tiles), all within one batch image.
// Per wave: A tile = 16 tokens x 32 dims held in 8 v2f WMMA A-fragments.
// Loop over codebook in 256-code LDS chunks; 16-code tiles -> 8x WMMA f32 16x16x4.
__global__ __launch_bounds__(TPB)
void vq_wmma_kernel(const float* __restrict__ x, const float* __restrict__ cb,
                    float* __restrict__ out, float* __restrict__ partial)
{
    __shared__ __align__(16) float lds_cb[CHUNK * LDS_STRIDE]; // staged codebook chunk
    __shared__ float lds_csq[CHUNK];                           // ||c||^2 for chunk
    __shared__ float lds_err[128];                             // per-token sq-err

    const int bid   = blockIdx.x;
    const int b     = bid >> 3;          // 8 blocks per image (1024/128)
    const int p0    = (bid & 7) * 128;   // base token position within image
    const int tid   = threadIdx.x;
    const int wv    = tid >> 5;
    const int lane  = tid & 31;
    const int lhalf = lane & 15;
    const bool hi   = lane >= 16;

    const float* xb = x + (size_t)b * (LATD * HW);

    // ---- Load A fragments: token M = lhalf, K-dims split across lane halves ----
    const int ptok = p0 + wv * 16 + lhalf;   // this lane's token row position
    v2f a[8];
#pragma unroll
    for (int i = 0; i < 8; ++i) {
        const int d0 = 4 * i + (hi ? 2 : 0);
        a[i].x = xb[d0 * HW + ptok];
        a[i].y = xb[(d0 + 1) * HW + ptok];
    }

    float best[8];
    int   bidx[8];
#pragma unroll
    for (int j = 0; j < 8; ++j) { best[j] = 3.4e38f; bidx[j] = 0; }

    // ---- Sweep codebook in LDS-staged chunks ----
    for (int ko = 0; ko < KCB; ko += CHUNK) {
        __syncthreads();   // protect LDS reuse from previous chunk
        // Stage CHUNK*32 contiguous floats via float4, re-laid with padded stride.
        const float4* src = (const float4*)(cb + (size_t)ko * LATD);
#pragma unroll
        for (int v = tid; v < CHUNK * 8; v += TPB) {
            float4 q = src[v];
            const int code = v >> 3;          // 8 float4 per code row
            const int dd   = (v & 7) * 4;
            float* dst = &lds_cb[code * LDS_STRIDE + dd];
            dst[0] = q.x; dst[1] = q.y; dst[2] = q.z; dst[3] = q.w;
        }
        __syncthreads();
        // Per-code squared norm (one code per thread; CHUNK == TPB).
        {
            float s = 0.f;
            const float* row = &lds_cb[tid * LDS_STRIDE];
#pragma unroll
            for (int d = 0; d < LATD; ++d) s += row[d] * row[d];
            lds_csq[tid] = s;
        }
        __syncthreads();

        // 16 code-tiles per chunk
        for (int t = 0; t < CHUNK / 16; ++t) {
            const float csq  = lds_csq[t * 16 + lhalf];
            const int   code = ko + t * 16 + lhalf;      // N = lhalf for this lane
            const float* bbase = &lds_cb[(t * 16 + lhalf) * LDS_STRIDE];

            v8f c = {};
#pragma unroll
            for (int i = 0; i < 8; ++i) {
                const int d0 = 4 * i + (hi ? 2 : 0);
                const v2f bb = *(const v2f*)(bbase + d0);
                // D = A(16x4) x B(4x16) + C ; accumulate dot over all 32 dims
                c = __builtin_amdgcn_wmma_f32_16x16x4_f32(
                        false, a[i], false, bb, (short)0, c, false, false);
            }
#pragma unroll
            for (int j = 0; j < 8; ++j) {
                const float m = csq - 2.0f * c[j];
                if (m < best[j]) { best[j] = m; bidx[j] = code; }  // strict <: first idx wins
            }
        }
    }

    // ---- Cross-lane argmin within each 16-lane half (rows 0-7 | 8-15) ----
#pragma unroll
    for (int j = 0; j < 8; ++j) {
        float bs = best[j];
        int   bi = bidx[j];
#pragma unroll
        for (int m = 8; m >= 1; m >>= 1) {
            const float os = __shfl_xor(bs, m, 32);
            const int   oi = __shfl_xor(bi, m, 32);
            if (os < bs || (os == bs && oi < bi)) { bs = os; bi = oi; }
        }
        best[j] = bs; bidx[j] = bi;
    }

    // ---- Winner writeback: lane lhalf=j owns token row j (+8 in upper half) ----
    float errsum = 0.f;
    if (lhalf < 8) {
        int myIdx = 0;
#pragma unroll
        for (int j = 0; j < 8; ++j) if (lhalf == j) myIdx = bidx[j];
        const int row   = lhalf + (hi ? 8 : 0);
        const int tok_p = p0 + wv * 16 + row;           // position within image
        const int n     = b * HW + tok_p;               // flat token id
        out[Q_ELEMS + n] = (float)myIdx;                // index output (as float)

        const float* crow = cb + (size_t)myIdx * LATD;  // L2-hot gather
#pragma unroll
        for (int d = 0; d < LATD; ++d) {
            const float q  = crow[d];
            out[(size_t)b * (LATD * HW) + d * HW + tok_p] = q;  // quant_out[b,d,h,w]
            const float e  = xb[d * HW + tok_p] - q;
            errsum += e * e;
        }
        lds_err[wv * 16 + row] = errsum;
    }
    __syncthreads();
    if (tid == 0) {
        float s = 0.f;
        for (int i = 0; i < 128; ++i) s += lds_err[i];   // fixed order: deterministic
        partial[bid] = s;
    }
}

// Deterministic final reduction of per-block partials -> both loss scalars.
__global__ void vq_loss_kernel(const float* __restrict__ partial, float* __restrict__ out)
{
    if (threadIdx.x == 0 && blockIdx.x == 0) {
        float s = 0.f;
        for (int i = 0; i < NTOK / 128; ++i) s += partial[i];
        const float mean = s / (float)(NTOK * LATD);
        out[Q_ELEMS + I_ELEMS]     = mean;  // codebook_loss
        out[Q_ELEMS + I_ELEMS + 1] = mean;  // commitment_loss (same forward value)
    }
}

extern "C" void kernel_launch(void* const* d_in, const int* in_sizes, int n_in,
                              void* d_out, int out_size, void* d_ws, size_t ws_size,
                              hipStream_t stream) {
    (void)in_sizes; (void)n_in; (void)out_size; (void)ws_size;
    const float* x  = (const float*)d_in[0];
    const float* cb = (const float*)d_in[1];
    float* out      = (float*)d_out;
    float* partial  = (float*)d_ws;   // 256 floats of scratch

    dim3 grid(NTOK / 128), block(TPB);
    vq_wmma_kernel<<<grid, block, 0, stream>>>(x, cb, out, partial);
    vq_loss_kernel<<<1, 32, 0, stream>>>(partial, out);
}